// GraphSequenceVAE_68247030333773
// MI455X (gfx1250) — compile-verified
//
#include <hip/hip_runtime.h>
#include <math.h>

#define JN    50
#define BATCH 32
#define TSEQ  256
#define HIDN  512
#define EMBD  256
#define LATD  256
#define POSE  150

typedef float v2f __attribute__((ext_vector_type(2)));
typedef float v8f __attribute__((ext_vector_type(8)));

// ---------------------------------------------------------------------------
// f32 WMMA wrapper: D(16x16) = A(16x4) x B(4x16) + C, wave32.
//   A lane layout: row m = lane&15, k = 2*(lane>>4) + r
//   B lane layout: col n = lane&15, k = 2*(lane>>4) + r
//   D lane layout: col n = lane&15, row m = vgpr + 8*(lane>>4)
// ---------------------------------------------------------------------------
__device__ __forceinline__ v8f wmma4(v2f a, v2f b, v8f c) {
  return __builtin_amdgcn_wmma_f32_16x16x4_f32(false, a, false, b, (short)0, c,
                                               false, false);
}

__device__ __forceinline__ float sigm(float x) { return 1.0f / (1.0f + expf(-x)); }

// ---------------------------------------------------------------------------
// Generic GEMM: C[M,N] = act(A[M,K] @ W[N,K]^T + bias[N])
// One wave (32 threads) per 16x16 output tile. ACT: 0=none, 1=exact GELU.
// ---------------------------------------------------------------------------
template <int ACT>
__global__ void gemm_wt(const float* __restrict__ A, const float* __restrict__ W,
                        const float* __restrict__ bias, float* __restrict__ C,
                        int M, int N, int K) {
  const int lane = threadIdx.x;
  const int col  = lane & 15;
  const int hi   = lane >> 4;
  const int n0 = blockIdx.x * 16;
  const int m0 = blockIdx.y * 16;
  const int arow = m0 + col;
  const int wrow = n0 + col;
  const bool rOk = arow < M;
  const bool wOk = wrow < N;
  v8f acc = {};
  for (int k = 0; k < K; k += 4) {
    const int ka = k + 2 * hi;
    v2f a, b;
    a.x = (rOk && ka     < K) ? A[(size_t)arow * K + ka]     : 0.f;
    a.y = (rOk && ka + 1 < K) ? A[(size_t)arow * K + ka + 1] : 0.f;
    b.x = (wOk && ka     < K) ? W[(size_t)wrow * K + ka]     : 0.f;
    b.y = (wOk && ka + 1 < K) ? W[(size_t)wrow * K + ka + 1] : 0.f;
    acc = wmma4(a, b, acc);
  }
  const int cN = n0 + col;
#pragma unroll
  for (int i = 0; i < 8; ++i) {
    const int r = m0 + i + 8 * hi;
    if (r < M && cN < N) {
      float v = acc[i] + bias[cN];
      if (ACT == 1) v = 0.5f * v * (1.0f + erff(v * 0.70710678118654752f));
      C[(size_t)r * N + cN] = v;
    }
  }
}

// ---------------------------------------------------------------------------
// GATv2 edge-softmax + aggregation, exploiting chain skeleton:
// node j's in-edges come from {j-1, j+1, j} (with bounds). One thread/node.
// y = elu(sum_k alpha_k * xl[nb_k] + bias)
// ---------------------------------------------------------------------------
__global__ void gat_aggregate(const float* __restrict__ xl, const float* __restrict__ xr,
                              const float* __restrict__ att, const float* __restrict__ bias,
                              float* __restrict__ y, int G, int F, int H, int C) {
  const int idx = blockIdx.x * blockDim.x + threadIdx.x;
  if (idx >= G * JN) return;
  const int g = idx / JN, j = idx % JN;
  int nb[3]; int nn = 0;
  if (j > 0)      nb[nn++] = j - 1;
  if (j < JN - 1) nb[nn++] = j + 1;
  nb[nn++] = j;  // self loop
  const float* xrj = xr + (size_t)idx * F;
  for (int h = 0; h < H; ++h) {
    const float* ah = att + h * C;
    float s[3];
    for (int k = 0; k < nn; ++k) {
      const float* xls = xl + ((size_t)g * JN + nb[k]) * F + h * C;
      float sc = 0.f;
      for (int c = 0; c < C; ++c) {
        float e = xls[c] + xrj[h * C + c];
        e = (e > 0.f) ? e : 0.2f * e;  // leaky_relu 0.2
        sc += ah[c] * e;
      }
      s[k] = sc;
    }
    float m = s[0];
    for (int k = 1; k < nn; ++k) m = fmaxf(m, s[k]);
    float w[3]; float den = 0.f;
    for (int k = 0; k < nn; ++k) { w[k] = expf(s[k] - m); den += w[k]; }
    const float inv = 1.f / den;
    for (int c = 0; c < C; ++c) {
      float acc = 0.f;
      for (int k = 0; k < nn; ++k)
        acc += w[k] * xl[((size_t)g * JN + nb[k]) * F + h * C + c];
      float v = acc * inv + bias[h * C + c];
      v = (v > 0.f) ? v : (expf(v) - 1.f);  // ELU
      y[(size_t)idx * F + h * C + c] = v;
    }
  }
}

// ---------------------------------------------------------------------------
// Bidirectional masked GRU layer (encoder). grid.x = direction (0 fwd, 1 bwd),
// 512 threads = 16 wave32. h[32x512] lives in LDS. Each wave owns 2 j-tiles of
// 16 hidden columns; per step it accumulates r/z/i/h gate tiles with WMMA over
// K=Din (input, from global) and K=512 (state, from LDS), then applies the
// pack_padded mask per batch row.
// ---------------------------------------------------------------------------
__global__ void gru_encoder_layer(
    const float* __restrict__ x, int xStrideT, int xStrideB, int Din,
    const float* __restrict__ Wi_f, const float* __restrict__ Wh_f,
    const float* __restrict__ bi_f, const float* __restrict__ bh_f,
    const float* __restrict__ Wi_b, const float* __restrict__ Wh_b,
    const float* __restrict__ bi_b, const float* __restrict__ bh_b,
    const unsigned char* __restrict__ mask,  // [B,T] bool
    float* __restrict__ ys, int ysStrideT, int ysStrideB,  // + dir*512 col off
    float* __restrict__ hT)  // [B, 1024], fwd cols 0:512, bwd 512:1024
{
  const int dir = blockIdx.x;
  const float* Wi = dir ? Wi_b : Wi_f;
  const float* Wh = dir ? Wh_b : Wh_f;
  const float* bi = dir ? bi_b : bi_f;
  const float* bh = dir ? bh_b : bh_f;
  __shared__ float hS[BATCH * HIDN];  // 64 KB
  for (int i = threadIdx.x; i < BATCH * HIDN; i += blockDim.x) hS[i] = 0.f;
  __syncthreads();
  const int lane = threadIdx.x & 31;
  const int wave = threadIdx.x >> 5;  // 0..15
  const int col  = lane & 15;
  const int hi   = lane >> 4;
  float hnew[2][2][8];
  for (int step = 0; step < TSEQ; ++step) {
    const int t = dir ? (TSEQ - 1 - step) : step;
    for (int jt = 0; jt < 2; ++jt) {
      const int jc = (wave * 2 + jt) * 16 + col;
      v8f accR[2] = {}, accZ[2] = {}, accI[2] = {}, accH[2] = {};
      // input part: gi = x_t @ Wi^T  (gates r,z,n)
      for (int k = 0; k < Din; k += 4) {
        const int kk = k + 2 * hi;
        const float* xa = x + (size_t)t * xStrideT + (size_t)col * xStrideB + kk;
        const float* xb = x + (size_t)t * xStrideT + (size_t)(16 + col) * xStrideB + kk;
        v2f aA = { xa[0], xa[1] };
        v2f aB = { xb[0], xb[1] };
        const float* wr = Wi + (size_t)jc * Din + kk;
        const float* wz = Wi + (size_t)(HIDN + jc) * Din + kk;
        const float* wn = Wi + (size_t)(2 * HIDN + jc) * Din + kk;
        v2f bR = { wr[0], wr[1] }, bZ = { wz[0], wz[1] }, bN = { wn[0], wn[1] };
        accR[0] = wmma4(aA, bR, accR[0]); accR[1] = wmma4(aB, bR, accR[1]);
        accZ[0] = wmma4(aA, bZ, accZ[0]); accZ[1] = wmma4(aB, bZ, accZ[1]);
        accI[0] = wmma4(aA, bN, accI[0]); accI[1] = wmma4(aB, bN, accI[1]);
      }
      // state part: gh = h @ Wh^T (r,z accumulate; n-gate kept separate)
      for (int k = 0; k < HIDN; k += 4) {
        const int kk = k + 2 * hi;
        v2f aA = { hS[col * HIDN + kk],        hS[col * HIDN + kk + 1] };
        v2f aB = { hS[(16 + col) * HIDN + kk], hS[(16 + col) * HIDN + kk + 1] };
        const float* wr = Wh + (size_t)jc * HIDN + kk;
        const float* wz = Wh + (size_t)(HIDN + jc) * HIDN + kk;
        const float* wn = Wh + (size_t)(2 * HIDN + jc) * HIDN + kk;
        v2f bR = { wr[0], wr[1] }, bZ = { wz[0], wz[1] }, bN = { wn[0], wn[1] };
        accR[0] = wmma4(aA, bR, accR[0]); accR[1] = wmma4(aB, bR, accR[1]);
        accZ[0] = wmma4(aA, bZ, accZ[0]); accZ[1] = wmma4(aB, bZ, accZ[1]);
        accH[0] = wmma4(aA, bN, accH[0]); accH[1] = wmma4(aB, bN, accH[1]);
      }
      const float bR = bi[jc] + bh[jc];
      const float bZ = bi[HIDN + jc] + bh[HIDN + jc];
      const float bI = bi[2 * HIDN + jc];
      const float bH = bh[2 * HIDN + jc];
      for (int mt = 0; mt < 2; ++mt) {
#pragma unroll
        for (int i = 0; i < 8; ++i) {
          const int b = mt * 16 + i + 8 * hi;
          const float r  = sigm(accR[mt][i] + bR);
          const float zz = sigm(accZ[mt][i] + bZ);
          const float n  = tanhf(accI[mt][i] + bI + r * (accH[mt][i] + bH));
          const float hold = hS[b * HIDN + jc];
          const float hv = (1.f - zz) * n + zz * hold;
          hnew[jt][mt][i] = mask[b * TSEQ + t] ? hv : hold;
        }
      }
    }
    __syncthreads();
    for (int jt = 0; jt < 2; ++jt) {
      const int jc = (wave * 2 + jt) * 16 + col;
      for (int mt = 0; mt < 2; ++mt) {
#pragma unroll
        for (int i = 0; i < 8; ++i) {
          const int b = mt * 16 + i + 8 * hi;
          const float hv = hnew[jt][mt][i];
          hS[b * HIDN + jc] = hv;
          ys[(size_t)t * ysStrideT + (size_t)b * ysStrideB + dir * HIDN + jc] = hv;
        }
      }
    }
    __syncthreads();
  }
  for (int i = threadIdx.x; i < BATCH * HIDN; i += blockDim.x) {
    const int b = i / HIDN, j = i % HIDN;
    hT[b * 2 * HIDN + dir * HIDN + j] = hS[i];
  }
}

// ---------------------------------------------------------------------------
// One GRU cell update for the decoder (state in LDS, input in LDS).
// ---------------------------------------------------------------------------
__device__ __forceinline__ void dec_gru_step(
    const float* __restrict__ xS, int Din, float* __restrict__ hS,
    const float* __restrict__ Wi, const float* __restrict__ Wh,
    const float* __restrict__ bi, const float* __restrict__ bh) {
  const int lane = threadIdx.x & 31;
  const int wave = threadIdx.x >> 5;
  const int col  = lane & 15;
  const int hi   = lane >> 4;
  float hnew[2][2][8];
  for (int jt = 0; jt < 2; ++jt) {
    const int jc = (wave * 2 + jt) * 16 + col;
    v8f accR[2] = {}, accZ[2] = {}, accI[2] = {}, accH[2] = {};
    for (int k = 0; k < Din; k += 4) {
      const int kk = k + 2 * hi;
      v2f aA = { xS[col * Din + kk],        xS[col * Din + kk + 1] };
      v2f aB = { xS[(16 + col) * Din + kk], xS[(16 + col) * Din + kk + 1] };
      const float* wr = Wi + (size_t)jc * Din + kk;
      const float* wz = Wi + (size_t)(HIDN + jc) * Din + kk;
      const float* wn = Wi + (size_t)(2 * HIDN + jc) * Din + kk;
      v2f bR = { wr[0], wr[1] }, bZ = { wz[0], wz[1] }, bN = { wn[0], wn[1] };
      accR[0] = wmma4(aA, bR, accR[0]); accR[1] = wmma4(aB, bR, accR[1]);
      accZ[0] = wmma4(aA, bZ, accZ[0]); accZ[1] = wmma4(aB, bZ, accZ[1]);
      accI[0] = wmma4(aA, bN, accI[0]); accI[1] = wmma4(aB, bN, accI[1]);
    }
    for (int k = 0; k < HIDN; k += 4) {
      const int kk = k + 2 * hi;
      v2f aA = { hS[col * HIDN + kk],        hS[col * HIDN + kk + 1] };
      v2f aB = { hS[(16 + col) * HIDN + kk], hS[(16 + col) * HIDN + kk + 1] };
      const float* wr = Wh + (size_t)jc * HIDN + kk;
      const float* wz = Wh + (size_t)(HIDN + jc) * HIDN + kk;
      const float* wn = Wh + (size_t)(2 * HIDN + jc) * HIDN + kk;
      v2f bR = { wr[0], wr[1] }, bZ = { wz[0], wz[1] }, bN = { wn[0], wn[1] };
      accR[0] = wmma4(aA, bR, accR[0]); accR[1] = wmma4(aB, bR, accR[1]);
      accZ[0] = wmma4(aA, bZ, accZ[0]); accZ[1] = wmma4(aB, bZ, accZ[1]);
      accH[0] = wmma4(aA, bN, accH[0]); accH[1] = wmma4(aB, bN, accH[1]);
    }
    const float bR = bi[jc] + bh[jc];
    const float bZ = bi[HIDN + jc] + bh[HIDN + jc];
    const float bI = bi[2 * HIDN + jc];
    const float bH = bh[2 * HIDN + jc];
    for (int mt = 0; mt < 2; ++mt) {
#pragma unroll
      for (int i = 0; i < 8; ++i) {
        const int b = mt * 16 + i + 8 * hi;
        const float r  = sigm(accR[mt][i] + bR);
        const float zz = sigm(accZ[mt][i] + bZ);
        const float n  = tanhf(accI[mt][i] + bI + r * (accH[mt][i] + bH));
        const float hold = hS[b * HIDN + jc];
        hnew[jt][mt][i] = (1.f - zz) * n + zz * hold;
      }
    }
  }
  __syncthreads();
  for (int jt = 0; jt < 2; ++jt) {
    const int jc = (wave * 2 + jt) * 16 + col;
    for (int mt = 0; mt < 2; ++mt) {
#pragma unroll
      for (int i = 0; i < 8; ++i) {
        const int b = mt * 16 + i + 8 * hi;
        hS[b * HIDN + jc] = hnew[jt][mt][i];
      }
    }
  }
  __syncthreads();
}

// ---------------------------------------------------------------------------
// Autoregressive 2-layer GRU decoder. Single block of 512 threads; dynamic
// LDS holds h0[32x512], h1[32x512], feedback input[32x256] (160 KB, within
// the 320 KB/WGP CDNA5 budget). pe written in [B,T,EMB] layout.
// ---------------------------------------------------------------------------
__global__ void gru_decoder(
    const float* __restrict__ h0init, const float* __restrict__ sos,
    const float* __restrict__ Wi0, const float* __restrict__ Wh0,
    const float* __restrict__ bi0, const float* __restrict__ bh0,
    const float* __restrict__ Wi1, const float* __restrict__ Wh1,
    const float* __restrict__ bi1, const float* __restrict__ bh1,
    const float* __restrict__ opW, const float* __restrict__ opB,
    float* __restrict__ pe) {
  extern __shared__ float sm[];
  float* h0S = sm;
  float* h1S = sm + BATCH * HIDN;
  float* inS = sm + 2 * BATCH * HIDN;
  for (int i = threadIdx.x; i < BATCH * HIDN; i += blockDim.x) {
    h0S[i] = h0init[i];
    h1S[i] = h0init[i];  // both layers share the latent-projected init
  }
  for (int i = threadIdx.x; i < BATCH * EMBD; i += blockDim.x)
    inS[i] = sos[i % EMBD];
  __syncthreads();
  const int lane = threadIdx.x & 31;
  const int wave = threadIdx.x >> 5;
  const int col  = lane & 15;
  const int hi   = lane >> 4;
  for (int t = 0; t < TSEQ; ++t) {
    dec_gru_step(inS, EMBD, h0S, Wi0, Wh0, bi0, bh0);
    dec_gru_step(h0S, HIDN, h1S, Wi1, Wh1, bi1, bh1);
    // out = h1 @ opW^T + opB ; feed back via inS, persist into pe[B,T,EMB]
    for (int p = 0; p < 2; ++p) {
      const int item = wave * 2 + p;   // 32 items = 16 n-tiles x 2 m-tiles
      const int n0 = (item >> 1) * 16;
      const int mt = item & 1;
      v8f acc = {};
      for (int k = 0; k < HIDN; k += 4) {
        const int kk = k + 2 * hi;
        v2f a = { h1S[(mt * 16 + col) * HIDN + kk],
                  h1S[(mt * 16 + col) * HIDN + kk + 1] };
        const float* w = opW + (size_t)(n0 + col) * HIDN + kk;
        v2f b = { w[0], w[1] };
        acc = wmma4(a, b, acc);
      }
      const int c = n0 + col;
#pragma unroll
      for (int i = 0; i < 8; ++i) {
        const int bb = mt * 16 + i + 8 * hi;
        const float v = acc[i] + opB[c];
        inS[bb * EMBD + c] = v;
        pe[((size_t)bb * TSEQ + t) * EMBD + c] = v;
      }
    }
    __syncthreads();
  }
}

// ---------------------------------------------------------------------------
// Small elementwise / row kernels
// ---------------------------------------------------------------------------
__global__ void reparam(const float* __restrict__ mu, const float* __restrict__ lv,
                        const float* __restrict__ eps, float* __restrict__ z, int n) {
  const int i = blockIdx.x * blockDim.x + threadIdx.x;
  if (i < n) z[i] = mu[i] + eps[i] * expf(0.5f * lv[i]);
}

__global__ void layernorm_rows(const float* __restrict__ X, const float* __restrict__ g,
                               const float* __restrict__ b, float* __restrict__ Y,
                               int M, int N) {
  const int r = blockIdx.x * blockDim.x + threadIdx.x;
  if (r >= M) return;
  const float* xr = X + (size_t)r * N;
  float mean = 0.f;
  for (int c = 0; c < N; ++c) mean += xr[c];
  mean /= (float)N;
  float var = 0.f;
  for (int c = 0; c < N; ++c) { const float d = xr[c] - mean; var += d * d; }
  var /= (float)N;
  const float inv = rsqrtf(var + 1e-5f);
  float* yr = Y + (size_t)r * N;
  for (int c = 0; c < N; ++c) yr[c] = (xr[c] - mean) * inv * g[c] + b[c];
}

// ---------------------------------------------------------------------------
// Host-side orchestration
// ---------------------------------------------------------------------------
extern "C" void kernel_launch(void* const* d_in, const int* in_sizes, int n_in,
                              void* d_out, int out_size, void* d_ws, size_t ws_size,
                              hipStream_t stream) {
  (void)in_sizes; (void)n_in; (void)out_size; (void)ws_size;
  const float* x = (const float*)d_in[0];                 // [B,T,150]
  const unsigned char* mask = (const unsigned char*)d_in[1];  // [B,T] bool
  const float* eps = (const float*)d_in[2];               // [B,256]
  // params in setup_inputs() insertion order, flattened depth-first
  const float* g0Wl = (const float*)d_in[3];
  const float* g0bl = (const float*)d_in[4];
  const float* g0Wr = (const float*)d_in[5];
  const float* g0br = (const float*)d_in[6];
  const float* g0att = (const float*)d_in[7];
  const float* g0bias = (const float*)d_in[8];
  const float* g1Wl = (const float*)d_in[9];
  const float* g1bl = (const float*)d_in[10];
  const float* g1Wr = (const float*)d_in[11];
  const float* g1br = (const float*)d_in[12];
  const float* g1att = (const float*)d_in[13];
  const float* g1bias = (const float*)d_in[14];
  const float* projW = (const float*)d_in[15];
  const float* projB = (const float*)d_in[16];
  const float* e00Wi = (const float*)d_in[17];
  const float* e00Wh = (const float*)d_in[18];
  const float* e00bi = (const float*)d_in[19];
  const float* e00bh = (const float*)d_in[20];
  const float* e01Wi = (const float*)d_in[21];
  const float* e01Wh = (const float*)d_in[22];
  const float* e01bi = (const float*)d_in[23];
  const float* e01bh = (const float*)d_in[24];
  const float* e10Wi = (const float*)d_in[25];
  const float* e10Wh = (const float*)d_in[26];
  const float* e10bi = (const float*)d_in[27];
  const float* e10bh = (const float*)d_in[28];
  const float* e11Wi = (const float*)d_in[29];
  const float* e11Wh = (const float*)d_in[30];
  const float* e11bi = (const float*)d_in[31];
  const float* e11bh = (const float*)d_in[32];
  const float* muW = (const float*)d_in[33];
  const float* mub = (const float*)d_in[34];
  const float* lvW = (const float*)d_in[35];
  const float* lvb = (const float*)d_in[36];
  const float* lpW = (const float*)d_in[37];
  const float* lpb = (const float*)d_in[38];
  const float* dc0Wi = (const float*)d_in[39];
  const float* dc0Wh = (const float*)d_in[40];
  const float* dc0bi = (const float*)d_in[41];
  const float* dc0bh = (const float*)d_in[42];
  const float* dc1Wi = (const float*)d_in[43];
  const float* dc1Wh = (const float*)d_in[44];
  const float* dc1bi = (const float*)d_in[45];
  const float* dc1bh = (const float*)d_in[46];
  const float* opW = (const float*)d_in[47];
  const float* opB = (const float*)d_in[48];
  const float* sos = (const float*)d_in[49];
  const float* m1W = (const float*)d_in[50];
  const float* m1b = (const float*)d_in[51];
  const float* lng = (const float*)d_in[52];
  const float* lnb = (const float*)d_in[53];
  const float* m2W = (const float*)d_in[54];
  const float* m2b = (const float*)d_in[55];

  float* out   = (float*)d_out;
  float* recon = out;                                    // [B,T,150]
  float* muO   = out + (size_t)BATCH * TSEQ * POSE;      // [B,256]
  float* lvO   = muO + (size_t)BATCH * LATD;             // [B,256]

  // workspace layout (linear bump allocator, ~177 MB of f32)
  float* ws = (float*)d_ws;
  size_t cur = 0;
  auto alloc = [&](size_t n) { float* p = ws + cur; cur += n; return p; };
  const int GC = 512;                   // graphs per GAT chunk
  const int NCH = (BATCH * TSEQ) / GC;  // 16 chunks
  float* xl0 = alloc((size_t)GC * JN * 64);
  float* xr0 = alloc((size_t)GC * JN * 64);
  float* y0  = alloc((size_t)GC * JN * 64);
  float* xl1 = alloc((size_t)GC * JN * 128);
  float* xr1 = alloc((size_t)GC * JN * 128);
  float* y1  = alloc((size_t)GC * JN * 128);
  float* emb = alloc((size_t)BATCH * TSEQ * EMBD);       // [B,T,256]
  float* xs1 = alloc((size_t)TSEQ * BATCH * 2 * HIDN);   // [T,B,1024]
  float* ys2 = alloc((size_t)TSEQ * BATCH * 2 * HIDN);   // scratch (layer1 ys)
  float* hT0 = alloc((size_t)BATCH * 2 * HIDN);          // scratch
  float* last = alloc((size_t)BATCH * 2 * HIDN);         // [B,1024]
  float* zb  = alloc((size_t)BATCH * LATD);
  float* h0i = alloc((size_t)BATCH * HIDN);
  float* pe  = alloc((size_t)BATCH * TSEQ * EMBD);       // [B,T,256]
  float* hm  = alloc((size_t)BATCH * TSEQ * 512);
  float* hm2 = alloc((size_t)BATCH * TSEQ * 512);

  // --- spatial GATv2 encoder + projection, chunked over graphs ---
  for (int c = 0; c < NCH; ++c) {
    const float* xc = x + (size_t)c * GC * POSE;  // [GC*J, 3] rows
    const int Mn = GC * JN;
    gemm_wt<0><<<dim3(64 / 16, Mn / 16), 32, 0, stream>>>(xc, g0Wl, g0bl, xl0, Mn, 64, 3);
    gemm_wt<0><<<dim3(64 / 16, Mn / 16), 32, 0, stream>>>(xc, g0Wr, g0br, xr0, Mn, 64, 3);
    gat_aggregate<<<(Mn + 255) / 256, 256, 0, stream>>>(xl0, xr0, g0att, g0bias, y0, GC, 64, 4, 16);
    gemm_wt<0><<<dim3(128 / 16, Mn / 16), 32, 0, stream>>>(y0, g1Wl, g1bl, xl1, Mn, 128, 64);
    gemm_wt<0><<<dim3(128 / 16, Mn / 16), 32, 0, stream>>>(y0, g1Wr, g1br, xr1, Mn, 128, 64);
    gat_aggregate<<<(Mn + 255) / 256, 256, 0, stream>>>(xl1, xr1, g1att, g1bias, y1, GC, 128, 4, 32);
    // y1 is [GC, 50*128] contiguous -> project to emb rows
    gemm_wt<0><<<dim3(EMBD / 16, GC / 16), 32, 0, stream>>>(
        y1, projW, projB, emb + (size_t)c * GC * EMBD, GC, EMBD, JN * 128);
  }

  // --- temporal bidirectional GRU encoder (2 layers) ---
  gru_encoder_layer<<<2, 512, 0, stream>>>(
      emb, EMBD, TSEQ * EMBD, EMBD,
      e00Wi, e00Wh, e00bi, e00bh, e01Wi, e01Wh, e01bi, e01bh,
      mask, xs1, BATCH * 2 * HIDN, 2 * HIDN, hT0);
  gru_encoder_layer<<<2, 512, 0, stream>>>(
      xs1, BATCH * 2 * HIDN, 2 * HIDN, 2 * HIDN,
      e10Wi, e10Wh, e10bi, e10bh, e11Wi, e11Wh, e11bi, e11bh,
      mask, ys2, BATCH * 2 * HIDN, 2 * HIDN, last);

  // --- latent head: mu / logvar straight into d_out; reparameterize ---
  gemm_wt<0><<<dim3(LATD / 16, BATCH / 16), 32, 0, stream>>>(last, muW, mub, muO, BATCH, LATD, 2 * HIDN);
  gemm_wt<0><<<dim3(LATD / 16, BATCH / 16), 32, 0, stream>>>(last, lvW, lvb, lvO, BATCH, LATD, 2 * HIDN);
  reparam<<<(BATCH * LATD + 255) / 256, 256, 0, stream>>>(muO, lvO, eps, zb, BATCH * LATD);
  gemm_wt<0><<<dim3(HIDN / 16, BATCH / 16), 32, 0, stream>>>(zb, lpW, lpb, h0i, BATCH, HIDN, LATD);

  // --- autoregressive GRU decoder (single persistent block, 160 KB LDS) ---
  const int DEC_LDS = (2 * BATCH * HIDN + BATCH * EMBD) * (int)sizeof(float);
  (void)hipFuncSetAttribute((const void*)gru_decoder,
                            hipFuncAttributeMaxDynamicSharedMemorySize, DEC_LDS);
  gru_decoder<<<1, 512, DEC_LDS, stream>>>(
      h0i, sos, dc0Wi, dc0Wh, dc0bi, dc0bh, dc1Wi, dc1Wh, dc1bi, dc1bh, opW, opB, pe);

  // --- spatial MLP decoder: Linear -> GELU -> LayerNorm -> Linear ---
  const int MT = BATCH * TSEQ;
  gemm_wt<1><<<dim3(512 / 16, MT / 16), 32, 0, stream>>>(pe, m1W, m1b, hm, MT, 512, EMBD);
  layernorm_rows<<<(MT + 255) / 256, 256, 0, stream>>>(hm, lng, lnb, hm2, MT, 512);
  gemm_wt<0><<<dim3((POSE + 15) / 16, MT / 16), 32, 0, stream>>>(hm2, m2W, m2b, recon, MT, POSE, 512);
}